// GCNBackbone_64321430225634
// MI455X (gfx1250) — compile-verified
//
#include <hip/hip_runtime.h>

// ---------------------------------------------------------------------------
// GCN backbone for MI455X (gfx1250, wave32).
// Bound by edge scatter (~3.4 GB L2/HBM traffic); GEMMs done with
// V_WMMA_F32_16X16X4_F32 (exact fp32 path, 16x16 tile per wave).
// ---------------------------------------------------------------------------

#define NNODES   100000
#define NEDGES   3200000
#define IN_DIM   128
#define HIDDEN   64
#define NGRAPHS  512

typedef __attribute__((ext_vector_type(2))) float v2f;
typedef __attribute__((ext_vector_type(8))) float v8f;

__device__ __forceinline__ void atomic_add_f32(float* p, float v) {
    // relaxed, agent scope -> lowers to global_atomic_add_f32 (L2 atomic unit)
    __hip_atomic_fetch_add(p, v, __ATOMIC_RELAXED, __HIP_MEMORY_SCOPE_AGENT);
}

// ---------------- degree / normalization -----------------------------------

__global__ void k_deg_init(float* dis) {
    int i = blockIdx.x * blockDim.x + threadIdx.x;
    if (i < NNODES) dis[i] = 1.0f;   // self-loop contributes 1 to in-degree
}

__global__ void k_deg_edges(const int* __restrict__ dst, float* __restrict__ dis) {
    int e = blockIdx.x * blockDim.x + threadIdx.x;
    if (e < NEDGES) atomic_add_f32(&dis[dst[e]], 1.0f);
}

__global__ void k_deg_rsqrt(float* dis) {
    int i = blockIdx.x * blockDim.x + threadIdx.x;
    if (i < NNODES) dis[i] = rsqrtf(dis[i]);   // deg >= 1 always
}

// ---------------- WMMA GEMM: hs = (X[N,K] @ W[K,64]) * dis[row] -------------
// One wave computes one 16x16 tile of the output. 6250*4 = 25000 waves,
// 8 waves per 256-thread block -> 3125 blocks, no remainder (EXEC all-1s).

template <int K>
__global__ __launch_bounds__(256) void k_gemm_wmma(const float* __restrict__ X,
                                                   const float* __restrict__ W,
                                                   const float* __restrict__ dis,
                                                   float* __restrict__ hs) {
    const int lane  = threadIdx.x & 31;
    const int wid   = (blockIdx.x * blockDim.x + threadIdx.x) >> 5;
    const int mtile = wid >> 2;            // 0..6249
    const int ntile = wid & 3;             // 0..3  (64 cols / 16)
    const int r     = lane & 15;           // A: row-in-tile, B: col-in-tile
    const int khalf = lane >> 4;           // K sub-block 0/1 per ISA layout

    const float* xrow = X + (size_t)(mtile * 16 + r) * K;
    const float* wcol = W + ntile * 16 + r;

    v8f c = {};
#pragma unroll
    for (int k0 = 0; k0 < K; k0 += 4) {
        // A (16x4, MxK): lanes 0-15 hold K={k0,k0+1}, lanes 16-31 K={k0+2,k0+3}
        v2f a = *(const v2f*)(xrow + k0 + 2 * khalf);
        // B (4x16, KxN): transpose-symmetric layout
        v2f b;
        b.x = wcol[(size_t)(k0 + 2 * khalf)     * HIDDEN];
        b.y = wcol[(size_t)(k0 + 2 * khalf + 1) * HIDDEN];
        c = __builtin_amdgcn_wmma_f32_16x16x4_f32(false, a, false, b,
                                                  (short)0, c, false, false);
    }

    // C/D layout: VGPR v -> row v (lanes 0-15) / v+8 (lanes 16-31), col = lane%16
#pragma unroll
    for (int v = 0; v < 8; ++v) {
        int row = mtile * 16 + khalf * 8 + v;
        hs[(size_t)row * HIDDEN + ntile * 16 + r] = c[v] * dis[row];
    }
}

// ---------------- self-loop init (acc = hs) ---------------------------------

__global__ void k_copy4(const float4* __restrict__ in, float4* __restrict__ out, int n4) {
    int i = blockIdx.x * blockDim.x + threadIdx.x;
    if (i < n4) out[i] = in[i];
}

// ---------------- edge scatter: acc[dst] += hs[src] -------------------------
// One wave per edge, 2 channels per lane (64 channels).

__global__ __launch_bounds__(256) void k_scatter(const float* __restrict__ hs,
                                                 float* __restrict__ acc,
                                                 const int* __restrict__ src,
                                                 const int* __restrict__ dst) {
    const int wid  = (blockIdx.x * blockDim.x + threadIdx.x) >> 5;
    const int lane = threadIdx.x & 31;
    if (wid >= NEDGES) return;                       // wave-uniform
    const int s = src[wid];
    const int d = dst[wid];
    float2 v = *(const float2*)(hs + (size_t)s * HIDDEN + lane * 2);
    float* o = acc + (size_t)d * HIDDEN + lane * 2;
    atomic_add_f32(o,     v.x);
    atomic_add_f32(o + 1, v.y);
}

// ---------------- epilogue: acc = relu(dis[i]*acc + b[ch]) ------------------

__global__ void k_epilogue(float* __restrict__ acc, const float* __restrict__ dis,
                           const float* __restrict__ bias) {
    int idx = blockIdx.x * blockDim.x + threadIdx.x;
    if (idx < NNODES * HIDDEN) {
        int i  = idx >> 6;
        int ch = idx & 63;
        float v = dis[i] * acc[idx] + bias[ch];
        acc[idx] = fmaxf(v, 0.0f);
    }
}

// ---------------- global mean pool ------------------------------------------

__global__ void k_pool_zero(float* __restrict__ out, float* __restrict__ cnt) {
    int idx = blockIdx.x * blockDim.x + threadIdx.x;
    if (idx < NGRAPHS * HIDDEN) out[idx] = 0.0f;
    if (idx < NGRAPHS)          cnt[idx] = 0.0f;
}

__global__ __launch_bounds__(256) void k_pool_acc(const float* __restrict__ h,
                                                  const int* __restrict__ batch,
                                                  float* __restrict__ out,
                                                  float* __restrict__ cnt) {
    const int wid  = (blockIdx.x * blockDim.x + threadIdx.x) >> 5;
    const int lane = threadIdx.x & 31;
    if (wid >= NNODES) return;                        // wave-uniform
    const int g = batch[wid];
    float2 v = *(const float2*)(h + (size_t)wid * HIDDEN + lane * 2);
    float* o = out + (size_t)g * HIDDEN + lane * 2;
    atomic_add_f32(o,     v.x);
    atomic_add_f32(o + 1, v.y);
    if (lane == 0) atomic_add_f32(&cnt[g], 1.0f);
}

__global__ void k_pool_div(float* __restrict__ out, const float* __restrict__ cnt) {
    int idx = blockIdx.x * blockDim.x + threadIdx.x;
    if (idx < NGRAPHS * HIDDEN) out[idx] /= fmaxf(cnt[idx >> 6], 1.0f);
}

// ---------------------------------------------------------------------------

extern "C" void kernel_launch(void* const* d_in, const int* in_sizes, int n_in,
                              void* d_out, int out_size, void* d_ws, size_t ws_size,
                              hipStream_t stream) {
    const float* x    = (const float*)d_in[0];
    const float* W1   = (const float*)d_in[1];
    const float* b1   = (const float*)d_in[2];
    const float* W2   = (const float*)d_in[3];
    const float* b2   = (const float*)d_in[4];
    const int*   ei   = (const int*)d_in[5];    // [2, E] row-major
    const int*   bat  = (const int*)d_in[6];
    const int*   src  = ei;
    const int*   dst  = ei + NEDGES;
    float* out = (float*)d_out;                 // [512, 64]

    float* buf0 = (float*)d_ws;                          // hs   : N x 64
    float* buf1 = buf0 + (size_t)NNODES * HIDDEN;        // acc  : N x 64
    float* dis  = buf1 + (size_t)NNODES * HIDDEN;        // N
    float* cnt  = dis + NNODES;                          // 512

    const int NH  = NNODES * HIDDEN;
    const int NH4 = NH / 4;

    // normalization coefficients dis = deg^{-1/2}
    k_deg_init <<<(NNODES + 255) / 256, 256, 0, stream>>>(dis);
    k_deg_edges<<<(NEDGES + 255) / 256, 256, 0, stream>>>(dst, dis);
    k_deg_rsqrt<<<(NNODES + 255) / 256, 256, 0, stream>>>(dis);

    // ---- layer 1 ----
    k_gemm_wmma<IN_DIM><<<3125, 256, 0, stream>>>(x, W1, dis, buf0);
    k_copy4   <<<(NH4 + 255) / 256, 256, 0, stream>>>((const float4*)buf0, (float4*)buf1, NH4);
    k_scatter <<<NEDGES / 8, 256, 0, stream>>>(buf0, buf1, src, dst);
    k_epilogue<<<(NH + 255) / 256, 256, 0, stream>>>(buf1, dis, b1);

    // ---- layer 2 ----
    k_gemm_wmma<HIDDEN><<<3125, 256, 0, stream>>>(buf1, W2, dis, buf0);
    k_copy4   <<<(NH4 + 255) / 256, 256, 0, stream>>>((const float4*)buf0, (float4*)buf1, NH4);
    k_scatter <<<NEDGES / 8, 256, 0, stream>>>(buf0, buf1, src, dst);
    k_epilogue<<<(NH + 255) / 256, 256, 0, stream>>>(buf1, dis, b2);

    // ---- global mean pool ----
    k_pool_zero<<<(NGRAPHS * HIDDEN + 255) / 256, 256, 0, stream>>>(out, cnt);
    k_pool_acc <<<NNODES / 8, 256, 0, stream>>>(buf1, bat, out, cnt);
    k_pool_div <<<(NGRAPHS * HIDDEN + 255) / 256, 256, 0, stream>>>(out, cnt);
}